// GCN_77893526880285
// MI455X (gfx1250) — compile-verified
//
#include <hip/hip_runtime.h>

typedef __attribute__((ext_vector_type(2))) float v2f;
typedef __attribute__((ext_vector_type(8))) float v8f;

// ---------------------------------------------------------------------------
// C[M,Nc] = A[M,K] @ B[K,Nc] in fp32 using V_WMMA_F32_16X16X4_F32.
// grid.x = M/16 (M-tile), blockDim.x = (Nc/16)*32: wave w owns N-tile w.
// Requires M%16==0, K%4==0, Nc%16==0 (true here: 100000/256/128/64).
// EXEC all-ones throughout (WMMA requirement) — no divergent guards.
// ---------------------------------------------------------------------------
__global__ void gcn_gemm_wmma_f32(const float* __restrict__ A,
                                  const float* __restrict__ B,
                                  float* __restrict__ C,
                                  int K, int Nc) {
  const int lane  = threadIdx.x & 31;
  const int wave  = threadIdx.x >> 5;
  const int mtile = blockIdx.x;
  const int half  = lane >> 4;            // 0 for lanes 0-15, 1 for 16-31
  const int idx16 = lane & 15;

  const int mrow  = (mtile << 4) + idx16; // A row handled by this lane
  const int ncol  = (wave  << 4) + idx16; // B/C column handled by this lane
  const int khalf = half << 1;            // K sub-offset: 0 or 2

  const float* Arow = A + (size_t)mrow * K;

  v8f acc = {};
  #pragma unroll 4
  for (int kk = 0; kk < K; kk += 4) {
    const int k0 = kk + khalf;
    // A 16x4 tile: lane holds A[m][k0], A[m][k0+1] (contiguous -> float2 load)
    v2f a = *(const v2f*)(Arow + k0);
    // B 4x16 tile: lane holds B[k0][n], B[k0+1][n]
    v2f b;
    b.x = B[(size_t)k0 * Nc + ncol];
    b.y = B[(size_t)(k0 + 1) * Nc + ncol];
    acc = __builtin_amdgcn_wmma_f32_16x16x4_f32(
        /*neg_a=*/false, a, /*neg_b=*/false, b,
        /*c_mod=*/(short)0, acc, /*reuse_a=*/false, /*reuse_b=*/false);
  }

  // D layout: VGPR v -> row (v + 8*half), col idx16
  const int mbase = (mtile << 4) + (half << 3);
  #pragma unroll
  for (int v = 0; v < 8; ++v) {
    C[(size_t)(mbase + v) * Nc + ncol] = acc[v];
  }
}

// ---------------------------------------------------------------------------
// Zero a float buffer (float4-vectorized, n4 = count of float4s).
// ---------------------------------------------------------------------------
__global__ void gcn_zero4(float4* __restrict__ p, long n4) {
  long i = (long)blockIdx.x * blockDim.x + threadIdx.x;
  if (i < n4) p[i] = make_float4(0.f, 0.f, 0.f, 0.f);
}

// ---------------------------------------------------------------------------
// Edge scatter, F=128: one wave per edge, float4 per lane (512B coalesced),
// f32 global atomics into agg[dst].
// ---------------------------------------------------------------------------
__global__ void gcn_scatter128(const float* __restrict__ sup,
                               const int* __restrict__ src,
                               const int* __restrict__ dst,
                               const float* __restrict__ ew,
                               float* __restrict__ agg, int E) {
  int e = blockIdx.x * (blockDim.x >> 5) + (threadIdx.x >> 5);
  if (e >= E) return;
  int lane = threadIdx.x & 31;
  int s = src[e], d = dst[e];
  float w = ew[e];
  const float4 v = *(const float4*)(sup + (size_t)s * 128 + lane * 4);
  float* o = agg + (size_t)d * 128 + lane * 4;
  atomicAdd(o + 0, v.x * w);
  atomicAdd(o + 1, v.y * w);
  atomicAdd(o + 2, v.z * w);
  atomicAdd(o + 3, v.w * w);
}

// Edge scatter, F=64: one wave per edge, float2 per lane (256B coalesced).
__global__ void gcn_scatter64(const float* __restrict__ sup,
                              const int* __restrict__ src,
                              const int* __restrict__ dst,
                              const float* __restrict__ ew,
                              float* __restrict__ agg, int E) {
  int e = blockIdx.x * (blockDim.x >> 5) + (threadIdx.x >> 5);
  if (e >= E) return;
  int lane = threadIdx.x & 31;
  int s = src[e], d = dst[e];
  float w = ew[e];
  const float2 v = *(const float2*)(sup + (size_t)s * 64 + lane * 2);
  float* o = agg + (size_t)d * 64 + lane * 2;
  atomicAdd(o + 0, v.x * w);
  atomicAdd(o + 1, v.y * w);
}

// ---------------------------------------------------------------------------
// out = relu(agg + bias[col]); F must be a power of two (128 here).
// ---------------------------------------------------------------------------
__global__ void gcn_bias_relu(const float* __restrict__ agg,
                              const float* __restrict__ bias,
                              float* __restrict__ out, long total, int Fmask) {
  long i = (long)blockIdx.x * blockDim.x + threadIdx.x;
  if (i >= total) return;
  float v = agg[i] + bias[(int)(i & Fmask)];
  out[i] = v > 0.f ? v : 0.f;
}

// ---------------------------------------------------------------------------
// Per-row (64 wide) bias + log_softmax; one wave per row, 2 cols per lane,
// wave32 shuffle reductions for max and sum.
// ---------------------------------------------------------------------------
__global__ void gcn_bias_logsoftmax64(const float* __restrict__ agg,
                                      const float* __restrict__ bias,
                                      float* __restrict__ out, int N) {
  int row = blockIdx.x * (blockDim.x >> 5) + (threadIdx.x >> 5);
  if (row >= N) return;
  int lane = threadIdx.x & 31;
  const float* a = agg + (size_t)row * 64;
  float x0 = a[lane]      + bias[lane];
  float x1 = a[lane + 32] + bias[lane + 32];
  float m = fmaxf(x0, x1);
  #pragma unroll
  for (int off = 16; off > 0; off >>= 1) m = fmaxf(m, __shfl_xor(m, off));
  float s = expf(x0 - m) + expf(x1 - m);
  #pragma unroll
  for (int off = 16; off > 0; off >>= 1) s += __shfl_xor(s, off);
  float lse = m + logf(s);
  float* o = out + (size_t)row * 64;
  o[lane]      = x0 - lse;
  o[lane + 32] = x1 - lse;
}

// ---------------------------------------------------------------------------
// Driver: layer1 GEMM -> scatter -> bias+relu (to out1),
//         layer2 GEMM (reads out1) -> scatter -> bias+log_softmax (to out2).
// ---------------------------------------------------------------------------
extern "C" void kernel_launch(void* const* d_in, const int* in_sizes, int n_in,
                              void* d_out, int out_size, void* d_ws, size_t ws_size,
                              hipStream_t stream) {
  const float* feature = (const float*)d_in[0];
  const int*   src     = (const int*)  d_in[1];
  const int*   dst     = (const int*)  d_in[2];
  const float* ew      = (const float*)d_in[3];
  const float* W1      = (const float*)d_in[4];
  const float* b1      = (const float*)d_in[5];
  const float* W2      = (const float*)d_in[6];
  const float* b2      = (const float*)d_in[7];

  const int H   = in_sizes[5];            // 128
  const int D   = in_sizes[7];            // 64
  const int Fin = in_sizes[4] / H;        // 256
  const int N   = in_sizes[0] / Fin;      // 100000
  const int E   = in_sizes[1];            // 1600000

  float* out1 = (float*)d_out;            // x1: [N, H]
  float* out2 = out1 + (size_t)N * H;     // log_softmax(x2): [N, D]

  float* sup = (float*)d_ws;              // support buffer, reused by both layers
  float* agg = sup + (size_t)N * H;       // aggregate buffer, reused

  const long totH = (long)N * H;
  const long totD = (long)N * D;
  const int  edgeBlocks = (E + 7) / 8;    // 8 waves (edges) per 256-thread block

  // ----- Layer 1 -----
  gcn_gemm_wmma_f32<<<N / 16, (H / 16) * 32, 0, stream>>>(feature, W1, sup, Fin, H);
  gcn_zero4<<<(int)((totH / 4 + 255) / 256), 256, 0, stream>>>((float4*)agg, totH / 4);
  gcn_scatter128<<<edgeBlocks, 256, 0, stream>>>(sup, src, dst, ew, agg, E);
  gcn_bias_relu<<<(int)((totH + 255) / 256), 256, 0, stream>>>(agg, b1, out1, totH, H - 1);

  // ----- Layer 2 (reads x1 from out1) -----
  gcn_gemm_wmma_f32<<<N / 16, (D / 16) * 32, 0, stream>>>(out1, W2, sup, H, D);
  gcn_zero4<<<(int)((totD / 4 + 255) / 256), 256, 0, stream>>>((float4*)agg, totD / 4);
  gcn_scatter64<<<edgeBlocks, 256, 0, stream>>>(sup, src, dst, ew, agg, E);
  gcn_bias_logsoftmax64<<<(N + 7) / 8, 256, 0, stream>>>(agg, b2, out2, N);
}